// GCNClassification_84808424227221
// MI455X (gfx1250) — compile-verified
//
#include <hip/hip_runtime.h>
#include <math.h>

typedef __attribute__((ext_vector_type(2))) float v2f;
typedef __attribute__((ext_vector_type(8))) float v8f;

// ---------------------------------------------------------------------------
// Degree / normalization
// ---------------------------------------------------------------------------
__global__ void init_deg_kernel(float* deg, int n) {
    int i = blockIdx.x * blockDim.x + threadIdx.x;
    if (i < n) deg[i] = 1.0f;  // self-loop
}

__global__ void deg_accum_kernel(const int* __restrict__ dst, float* deg, int e) {
    int i = blockIdx.x * blockDim.x + threadIdx.x;
    if (i < e) atomicAdd(&deg[dst[i]], 1.0f);
}

__global__ void rsqrt_kernel(float* d, int n) {
    int i = blockIdx.x * blockDim.x + threadIdx.x;
    if (i < n) d[i] = rsqrtf(d[i]);
}

__global__ void norm_kernel(const int* __restrict__ src, const int* __restrict__ dst,
                            const float* __restrict__ dis, float* __restrict__ nrm, int e) {
    int i = blockIdx.x * blockDim.x + threadIdx.x;
    if (i < e) nrm[i] = dis[src[i]] * dis[dst[i]];
}

// ---------------------------------------------------------------------------
// GEMM1: h1[N,128] = x[N,128] @ W1[128,128]   (f32 WMMA 16x16x4)
// 8 waves per block; wave w owns output columns [16w, 16w+16); block owns a
// 16-row stripe (grid-stride). W1 staged in LDS; B-frags preloaded to regs.
// ---------------------------------------------------------------------------
__global__ void gemm1_kernel(const float* __restrict__ x,
                             const float* __restrict__ W,
                             float* __restrict__ h, int nstripes) {
    __shared__ float lW[128 * 128];  // 64 KB
    const int tid  = threadIdx.x;
    const int wave = tid >> 5;
    const int lane = tid & 31;
    for (int i = tid; i < 128 * 128; i += 256) lW[i] = W[i];
    __syncthreads();

    const int n0 = wave * 16;
    const int lm = lane & 15;           // col within tile / row within stripe
    const int kh = (lane >> 4) * 2;     // K sub-offset: 0 or 2

    // B fragment for k-step t: element (k, n) at lane = n + 16*(k>>1), vgpr = k&1
    v2f bf[32];
#pragma unroll
    for (int t = 0; t < 32; ++t) {
        int k = 4 * t + kh;
        bf[t].x = lW[(k + 0) * 128 + n0 + lm];
        bf[t].y = lW[(k + 1) * 128 + n0 + lm];
    }

    for (int s = blockIdx.x; s < nstripes; s += gridDim.x) {
        const int row0 = s * 16;
        const float* xr = x + (size_t)(row0 + lm) * 128 + kh;
        v8f acc = {};
#pragma unroll
        for (int t = 0; t < 32; ++t) {
            v2f a;
            a.x = xr[4 * t + 0];
            a.y = xr[4 * t + 1];
            acc = __builtin_amdgcn_wmma_f32_16x16x4_f32(
                false, a, false, bf[t], (short)0, acc, false, false);
        }
        const int mhi = (lane >> 4) * 8;
#pragma unroll
        for (int vr = 0; vr < 8; ++vr)
            h[(size_t)(row0 + mhi + vr) * 128 + n0 + lm] = acc[vr];
    }
}

// ---------------------------------------------------------------------------
// GEMM2: h2[N,16] = z[N,128] @ W2[128,16]   (single 16-col tile, wave/stripe)
// ---------------------------------------------------------------------------
__global__ void gemm2_kernel(const float* __restrict__ z,
                             const float* __restrict__ W,
                             float* __restrict__ h, int nstripes) {
    __shared__ float lW[128 * 16];
    const int tid  = threadIdx.x;
    const int wave = tid >> 5;
    const int lane = tid & 31;
    for (int i = tid; i < 128 * 16; i += 256) lW[i] = W[i];
    __syncthreads();

    const int lm = lane & 15;
    const int kh = (lane >> 4) * 2;

    v2f bf[32];
#pragma unroll
    for (int t = 0; t < 32; ++t) {
        int k = 4 * t + kh;
        bf[t].x = lW[(k + 0) * 16 + lm];
        bf[t].y = lW[(k + 1) * 16 + lm];
    }

    for (int s = blockIdx.x * 8 + wave; s < nstripes; s += gridDim.x * 8) {
        const int row0 = s * 16;
        const float* zr = z + (size_t)(row0 + lm) * 128 + kh;
        v8f acc = {};
#pragma unroll
        for (int t = 0; t < 32; ++t) {
            v2f a;
            a.x = zr[4 * t + 0];
            a.y = zr[4 * t + 1];
            acc = __builtin_amdgcn_wmma_f32_16x16x4_f32(
                false, a, false, bf[t], (short)0, acc, false, false);
        }
        const int mhi = (lane >> 4) * 8;
#pragma unroll
        for (int vr = 0; vr < 8; ++vr)
            h[(size_t)(row0 + mhi + vr) * 16 + lm] = acc[vr];
    }
}

// ---------------------------------------------------------------------------
// Aggregation: init with self-loop term (also serves as the zero-init), then
// wave-per-edge atomic scatter-add of norm-scaled source rows.
// ---------------------------------------------------------------------------
template <int F>
__global__ void agg_init_kernel(const float* __restrict__ h,
                                const float* __restrict__ dis,
                                float* __restrict__ agg, int n) {
    int i = blockIdx.x * blockDim.x + threadIdx.x;
    if (i < n * F) {
        float d = dis[i / F];
        agg[i] = h[i] * d * d;
    }
}

template <int F>
__global__ void agg_edge_kernel(const float* __restrict__ h,
                                const float* __restrict__ nrm,
                                const int* __restrict__ src,
                                const int* __restrict__ dst,
                                float* __restrict__ agg, int e) {
    int w    = (blockIdx.x * blockDim.x + threadIdx.x) >> 5;
    int lane = threadIdx.x & 31;
    if (w >= e) return;
    const int   s  = src[w];
    const int   d  = dst[w];
    const float nm = nrm[w];
    if (F == 128) {
        const float4 v = ((const float4*)(h + (size_t)s * 128))[lane];
        float* ad = agg + (size_t)d * 128 + 4 * lane;
        atomicAdd(ad + 0, v.x * nm);
        atomicAdd(ad + 1, v.y * nm);
        atomicAdd(ad + 2, v.z * nm);
        atomicAdd(ad + 3, v.w * nm);
    } else {
        if (lane < F)
            atomicAdd(agg + (size_t)d * F + lane, h[(size_t)s * F + lane] * nm);
    }
}

__global__ void bias_relu_kernel(float* __restrict__ a, const float* __restrict__ b, int n) {
    int i = blockIdx.x * blockDim.x + threadIdx.x;
    if (i < n * 128) {
        float v = a[i] + b[i & 127];
        a[i] = v > 0.0f ? v : 0.0f;
    }
}

// ---------------------------------------------------------------------------
// Pooling + log_softmax
// ---------------------------------------------------------------------------
__global__ void zero_pool_kernel(float* pooled, float* counts, int g) {
    int i = blockIdx.x * blockDim.x + threadIdx.x;
    if (i < g * 16) pooled[i] = 0.0f;
    if (i < g) counts[i] = 0.0f;
}

__global__ void pool_accum_kernel(const float* __restrict__ h2,
                                  const int* __restrict__ batch,
                                  float* pooled, float* counts, int n) {
    int i = blockIdx.x * blockDim.x + threadIdx.x;
    if (i < n * 16) {
        int node = i >> 4, f = i & 15;
        int g = batch[node];
        atomicAdd(&pooled[g * 16 + f], h2[i]);
        if (f == 0) atomicAdd(&counts[g], 1.0f);
    }
}

__global__ void logsoftmax_kernel(const float* __restrict__ pooled,
                                  const float* __restrict__ counts,
                                  const float* __restrict__ b2,
                                  float* __restrict__ out, int g) {
    int i = blockIdx.x * blockDim.x + threadIdx.x;
    if (i >= g) return;
    float inv = 1.0f / fmaxf(counts[i], 1.0f);
    float v[16];
    float mx = -INFINITY;
#pragma unroll
    for (int f = 0; f < 16; ++f) {
        v[f] = pooled[i * 16 + f] * inv + b2[f];
        mx = fmaxf(mx, v[f]);
    }
    float s = 0.0f;
#pragma unroll
    for (int f = 0; f < 16; ++f) s += expf(v[f] - mx);
    float ls = logf(s);
#pragma unroll
    for (int f = 0; f < 16; ++f) out[i * 16 + f] = v[f] - mx - ls;
}

// ---------------------------------------------------------------------------
extern "C" void kernel_launch(void* const* d_in, const int* in_sizes, int n_in,
                              void* d_out, int out_size, void* d_ws, size_t ws_size,
                              hipStream_t stream) {
    const float* x    = (const float*)d_in[0];
    const float* W1   = (const float*)d_in[1];
    const float* b1   = (const float*)d_in[2];
    const float* W2   = (const float*)d_in[3];
    const float* b2   = (const float*)d_in[4];
    const int* esrc   = (const int*)d_in[5];
    const int* edst   = (const int*)d_in[6];
    const int* batch  = (const int*)d_in[7];
    float* out        = (float*)d_out;

    const int N = in_sizes[0] / 128;
    const int E = in_sizes[5];
    const int G = out_size / 16;
    const int Ep = (E + 3) & ~3;  // keep h1 16B-aligned

    float* ws     = (float*)d_ws;
    float* dis    = ws;                            // N
    float* nrm    = dis + N;                       // Ep
    float* h1     = nrm + Ep;                      // N*128
    float* agg1   = h1 + (size_t)N * 128;          // N*128   (z1 after relu)
    float* h2     = agg1 + (size_t)N * 128;        // N*16
    float* agg2   = h2 + (size_t)N * 16;           // N*16
    float* pooled = agg2 + (size_t)N * 16;         // G*16
    float* counts = pooled + (size_t)G * 16;       // G

    const int B = 256;
    const int nstripes = (N + 15) / 16;

    // degree / norm
    init_deg_kernel<<<(N + B - 1) / B, B, 0, stream>>>(dis, N);
    deg_accum_kernel<<<(E + B - 1) / B, B, 0, stream>>>(edst, dis, E);
    rsqrt_kernel<<<(N + B - 1) / B, B, 0, stream>>>(dis, N);
    norm_kernel<<<(E + B - 1) / B, B, 0, stream>>>(esrc, edst, dis, nrm, E);

    // layer 1
    gemm1_kernel<<<nstripes, B, 0, stream>>>(x, W1, h1, nstripes);
    agg_init_kernel<128><<<((size_t)N * 128 + B - 1) / B, B, 0, stream>>>(h1, dis, agg1, N);
    agg_edge_kernel<128><<<((size_t)E * 32 + B - 1) / B, B, 0, stream>>>(h1, nrm, esrc, edst, agg1, E);
    bias_relu_kernel<<<((size_t)N * 128 + B - 1) / B, B, 0, stream>>>(agg1, b1, N);

    // layer 2
    gemm2_kernel<<<(nstripes + 7) / 8, B, 0, stream>>>(agg1, W2, h2, nstripes);
    agg_init_kernel<16><<<((size_t)N * 16 + B - 1) / B, B, 0, stream>>>(h2, dis, agg2, N);
    agg_edge_kernel<16><<<((size_t)E * 32 + B - 1) / B, B, 0, stream>>>(h2, nrm, esrc, edst, agg2, E);

    // pooling + log_softmax
    zero_pool_kernel<<<(G * 16 + B - 1) / B, B, 0, stream>>>(pooled, counts, G);
    pool_accum_kernel<<<((size_t)N * 16 + B - 1) / B, B, 0, stream>>>(agg2, batch, pooled, counts, N);
    logsoftmax_kernel<<<(G + B - 1) / B, B, 0, stream>>>(pooled, counts, b2, out, G);
}